// SensoryCortex42PyTorch_87935160418550
// MI455X (gfx1250) — compile-verified
//
#include <hip/hip_runtime.h>
#include <stdint.h>

// ---------------------------------------------------------------------------
// SensoryCortex gather + competitive normalization, MI455X (gfx1250).
//
// out[b,n] = max(enc[b, pref[n] mod F] * w[n] - 0.1 * mean_n(...), 0)
//
// Memory-bound: ~512 MiB of HBM traffic (read enc once, write out once)
// -> ~22 us floor at 23.3 TB/s. Strategy: stage each 64 KiB enc row into LDS
// with CDNA5 async global->LDS copies (ASYNCcnt), gather from LDS, one block
// per row, register-staged values between the reduction and the store pass.
// ---------------------------------------------------------------------------

#define BLOCK 1024
#define MAX_LDS_F 16384   // 64 KiB of fp32 row data

typedef float  float4_t __attribute__((ext_vector_type(4)));
typedef int    int4_t   __attribute__((ext_vector_type(4)));
typedef __attribute__((address_space(1))) int4_t as1_int4;   // global v4i
typedef __attribute__((address_space(3))) int4_t as3_int4;   // LDS v4i

template <int CHUNKS>
__global__ __launch_bounds__(BLOCK)
void sensory_cortex_kernel(const float* __restrict__ enc,   // [B, F]
                           const float* __restrict__ w,     // [N]
                           const int*   __restrict__ pref,  // [N]
                           float* __restrict__ out,         // [B, N]
                           int N, int F)
{
    __shared__ float buf[MAX_LDS_F];        // staged encoded_features row

    const int b   = blockIdx.x;
    const int tid = threadIdx.x;
    const float* enc_row = enc + (size_t)b * (size_t)F;

    // ---- Stage the row into LDS (16 bytes per lane per issue) -------------
#if __has_builtin(__builtin_amdgcn_global_load_async_to_lds_b128)
    for (int i = tid * 4; i < F; i += BLOCK * 4) {
        // int->pointer casts dodge addrspace conversion checks; generic LDS
        // addresses carry the LDS byte offset in the low 32 bits.
        as1_int4* gp = (as1_int4*)(uintptr_t)(enc_row + i);
        as3_int4* lp = (as3_int4*)(uintptr_t)(&buf[i]);
        __builtin_amdgcn_global_load_async_to_lds_b128(gp, lp,
                                                       /*offset=*/0,
                                                       /*cpol=*/0);
    }
  #if __has_builtin(__builtin_amdgcn_s_wait_asynccnt)
    __builtin_amdgcn_s_wait_asynccnt(0);
  #else
    asm volatile("s_wait_asynccnt 0" ::: "memory");
  #endif
#else
    // Fallback: vector load + LDS store (global_load_b128 + ds_store_b128)
    for (int i = tid * 4; i < F; i += BLOCK * 4) {
        float4_t v = __builtin_nontemporal_load((const float4_t*)(enc_row + i));
        *(float4_t*)(&buf[i]) = v;
    }
#endif
    __syncthreads();

    // ---- Gather + specialize; stage results in registers; partial sum -----
    float spec[CHUNKS * 4];
    float psum = 0.0f;
#pragma unroll
    for (int c = 0; c < CHUNKS; ++c) {
        const int n0 = tid * 4 + c * (BLOCK * 4);
        if (n0 < N) {
            int4_t   p4 = *(const int4_t*)(pref + n0);
            float4_t w4 = *(const float4_t*)(w + n0);
#pragma unroll
            for (int j = 0; j < 4; ++j) {
                int idx = p4[j] % F;          // jnp.mod semantics
                if (idx < 0) idx += F;
                float s = buf[idx] * w4[j];
                spec[c * 4 + j] = s;
                psum += s;
            }
        }
    }

    // ---- Block reduction of psum (wave32 shuffle tree, then cross-wave) ---
#pragma unroll
    for (int off = 16; off > 0; off >>= 1)
        psum += __shfl_down(psum, off, 32);

    __syncthreads();                       // all LDS gathers done -> reuse buf
    if ((tid & 31) == 0) buf[tid >> 5] = psum;   // wave sums in buf[0..31]
    __syncthreads();

    if (tid < 32) {
        constexpr int NWAVES = BLOCK / 32;
        float s = (tid < NWAVES) ? buf[tid] : 0.0f;
#pragma unroll
        for (int off = 16; off > 0; off >>= 1)
            s += __shfl_down(s, off, 32);
        if (tid == 0) buf[0] = s;
    }
    __syncthreads();

    const float bias = 0.1f * (buf[0] / (float)N);   // MEAN_SUB * mean

    // ---- Subtract, rectify, stream out (non-temporal: written once) ------
    float* out_row = out + (size_t)b * (size_t)N;
#pragma unroll
    for (int c = 0; c < CHUNKS; ++c) {
        const int n0 = tid * 4 + c * (BLOCK * 4);
        if (n0 < N) {
            float4_t o;
#pragma unroll
            for (int j = 0; j < 4; ++j)
                o[j] = fmaxf(spec[c * 4 + j] - bias, 0.0f);
            __builtin_nontemporal_store(o, (float4_t*)(out_row + n0));
        }
    }
}

template <int CHUNKS>
static void launch_sc(const float* enc, const float* w, const int* pref,
                      float* out, int B, int N, int F, hipStream_t stream)
{
    sensory_cortex_kernel<CHUNKS><<<dim3(B), dim3(BLOCK), 0, stream>>>(
        enc, w, pref, out, N, F);
}

extern "C" void kernel_launch(void* const* d_in, const int* in_sizes, int n_in,
                              void* d_out, int out_size, void* d_ws, size_t ws_size,
                              hipStream_t stream)
{
    (void)n_in; (void)d_ws; (void)ws_size;
    const float* enc  = (const float*)d_in[0];   // [B, F] fp32
    const float* w    = (const float*)d_in[1];   // [N]    fp32
    const int*   pref = (const int*)d_in[2];     // [N]    int32
    float* out = (float*)d_out;                  // [B, N] fp32

    const int N = in_sizes[1];
    const int B = out_size / N;
    const int F = in_sizes[0] / B;               // 16384 -> fits 64 KiB LDS

    const int chunks = (N + BLOCK * 4 - 1) / (BLOCK * 4);
    switch (chunks) {
        case 1: launch_sc<1>(enc, w, pref, out, B, N, F, stream); break;
        case 2: launch_sc<2>(enc, w, pref, out, B, N, F, stream); break;
        case 3: launch_sc<3>(enc, w, pref, out, B, N, F, stream); break;
        case 4: launch_sc<4>(enc, w, pref, out, B, N, F, stream); break;
        case 5: launch_sc<5>(enc, w, pref, out, B, N, F, stream); break;
        case 6: launch_sc<6>(enc, w, pref, out, B, N, F, stream); break;
        case 7: launch_sc<7>(enc, w, pref, out, B, N, F, stream); break;
        default: launch_sc<8>(enc, w, pref, out, B, N, F, stream); break;
    }
}